// OptimalNet_13743895347676
// MI455X (gfx1250) — compile-verified
//
#include <hip/hip_runtime.h>

typedef __attribute__((ext_vector_type(2))) float v2f;
typedef __attribute__((ext_vector_type(8))) float v8f;
typedef __attribute__((ext_vector_type(4))) int   v4i;

#define B_ROWS 1024
#define DIM    4096
#define NSLAB  (DIM / 32)

// gfx1250 async global->LDS staging (ASYNCcnt-tracked), with portable fallback.
#if defined(__gfx1250__) && __has_builtin(__builtin_amdgcn_global_load_async_to_lds_b128) && __has_builtin(__builtin_amdgcn_s_wait_asynccnt)
#define USE_ASYNC_LDS 1
#else
#define USE_ASYNC_LDS 0
#endif

// ---------------------------------------------------------------------------
// vj = relu(vi @ W^T + bias)
// C[b,o] = sum_k vi[b,k] * W[o,k]  (both K-contiguous, row-major).
// Block tile 128(M) x 128(N), 256 threads = 8 waves (4 M-waves x 2 N-waves),
// each wave computes 32x64 = 2x4 f32 16x16x4 WMMA accumulators.
// LDS double-buffered; tiles streamed in with GLOBAL_LOAD_ASYNC_TO_LDS_B128.
// ---------------------------------------------------------------------------
__global__ __launch_bounds__(256)
void gemm_relu_wmma(const float* __restrict__ vi, const float* __restrict__ W,
                    const float* __restrict__ bias, float* __restrict__ vj)
{
    // +4 pad: row stride 36 floats = 144B (16B-aligned float4s, conflict-free
    // 16-row fragment gathers: gcd(36,64)=4 -> distinct banks, halves disjoint)
    __shared__ float As[2][128][36];
    __shared__ float Ws[2][128][36];

    const int t     = threadIdx.x;
    const int lane  = t & 31;
    const int wave  = t >> 5;            // 0..7
    const int l16   = lane & 15;         // M/N index within fragment
    const int lhalf = lane >> 4;         // K-pair select (ISA A/B layout)
    const int wm    = (wave & 3) * 32;   // wave M offset in block tile
    const int wn    = (wave >> 2) * 64;  // wave N offset in block tile
    const int m0    = blockIdx.y * 128;
    const int n0    = blockIdx.x * 128;

    v8f acc[2][4] = {};

    // Stage one 128x32 A tile + 128x32 W tile into LDS buffer `b`.
    // 256 threads x 4 iters x (1 A + 1 W) b128 = whole slab; 8 async ops/thread.
    auto issue_slab = [&](int s, int b) {
        const int k0 = s * 32;
        #pragma unroll
        for (int i = 0; i < 4; ++i) {
            const int j  = i * 256 + t;
            const int r  = j >> 3;
            const int c4 = (j & 7) << 2;
            const float* ga = &vi[(size_t)(m0 + r) * DIM + k0 + c4];
            const float* gw = &W[(size_t)(n0 + r) * DIM + k0 + c4];
#if USE_ASYNC_LDS
            __builtin_amdgcn_global_load_async_to_lds_b128(
                (v4i*)ga, (v4i*)&As[b][r][c4], 0, 0);
            __builtin_amdgcn_global_load_async_to_lds_b128(
                (v4i*)gw, (v4i*)&Ws[b][r][c4], 0, 0);
#else
            *(float4*)&As[b][r][c4] = *(const float4*)ga;
            *(float4*)&Ws[b][r][c4] = *(const float4*)gw;
#endif
        }
    };

    issue_slab(0, 0);

    for (int s = 0; s < NSLAB; ++s) {
        const int cur = s & 1;
        // Prefetch next slab into the other buffer (its readers finished at
        // the trailing barrier of iteration s-1).
        if (s + 1 < NSLAB) issue_slab(s + 1, cur ^ 1);
#if USE_ASYNC_LDS
        // Async loads complete in order: waiting down to the 8 just-issued
        // ops guarantees slab s's transfers have landed in LDS.
        if (s + 1 < NSLAB) __builtin_amdgcn_s_wait_asynccnt(8);
        else               __builtin_amdgcn_s_wait_asynccnt(0);
#endif
        __syncthreads();   // all waves' slab-s data visible

        #pragma unroll
        for (int kk = 0; kk < 32; kk += 4) {
            const int ka = kk + lhalf * 2;
            v2f a[2], b[4];
            #pragma unroll
            for (int tm = 0; tm < 2; ++tm) {
                a[tm][0] = As[cur][wm + tm * 16 + l16][ka];
                a[tm][1] = As[cur][wm + tm * 16 + l16][ka + 1];
            }
            #pragma unroll
            for (int tn = 0; tn < 4; ++tn) {
                b[tn][0] = Ws[cur][wn + tn * 16 + l16][ka];
                b[tn][1] = Ws[cur][wn + tn * 16 + l16][ka + 1];
            }
            #pragma unroll
            for (int tm = 0; tm < 2; ++tm)
                #pragma unroll
                for (int tn = 0; tn < 4; ++tn)
                    acc[tm][tn] = __builtin_amdgcn_wmma_f32_16x16x4_f32(
                        false, a[tm], false, b[tn], (short)0, acc[tm][tn],
                        false, false);
        }
        __syncthreads();   // readers done before buffer is overwritten
    }

    // Epilogue: bias + relu + store.
    // C/D layout: VGPR r -> M = r (lanes 0-15) / r+8 (lanes 16-31), N = l16.
    #pragma unroll
    for (int tm = 0; tm < 2; ++tm) {
        #pragma unroll
        for (int tn = 0; tn < 4; ++tn) {
            const int gn   = n0 + wn + tn * 16 + l16;
            const float bb = bias[gn];
            const int gmb  = m0 + wm + tm * 16 + lhalf * 8;
            #pragma unroll
            for (int r = 0; r < 8; ++r) {
                float v = acc[tm][tn][r] + bb;
                v = v > 0.0f ? v : 0.0f;
                vj[(size_t)(gmb + r) * DIM + gn] = v;
            }
        }
    }
}

// Stage 1: partial[rb][o] = sum over 128 rows of vj[:, o]
__global__ __launch_bounds__(256)
void colsum_part(const float* __restrict__ vj, float* __restrict__ partial)
{
    const int o  = blockIdx.x * 256 + threadIdx.x;
    const int rb = blockIdx.y;             // 0..7
    float s = 0.0f;
    const int b0 = rb * 128;
    for (int b = b0; b < b0 + 128; ++b)
        s += vj[(size_t)b * DIM + o];
    partial[(size_t)rb * DIM + o] = s;
}

// Stage 2: shift[o] = (HEBB*RATE/batch_num/B) * colsum[o]
__global__ __launch_bounds__(256)
void colsum_final(const float* __restrict__ partial, float* __restrict__ shift,
                  const int* __restrict__ bn)
{
    const int o = blockIdx.x * 256 + threadIdx.x;
    float s = 0.0f;
    #pragma unroll
    for (int r = 0; r < 8; ++r)
        s += partial[(size_t)r * DIM + o];
    const float c = (0.01f * 0.001f) / (float)(*bn);
    shift[o] = s * (c / (float)B_ROWS);
}

// rowsum[b] = sum_i vi[b,i]
__global__ __launch_bounds__(256)
void rowsum_k(const float* __restrict__ vi, float* __restrict__ rowsum)
{
    __shared__ float red[256];
    const int b = blockIdx.x;
    float s = 0.0f;
    for (int i = threadIdx.x; i < DIM; i += 256)
        s += vi[(size_t)b * DIM + i];
    red[threadIdx.x] = s;
    __syncthreads();
    for (int off = 128; off > 0; off >>= 1) {
        if (threadIdx.x < off) red[threadIdx.x] += red[threadIdx.x + off];
        __syncthreads();
    }
    if (threadIdx.x == 0) rowsum[b] = red[0];
}

// In-place: vj[b,o] += rowsum[b] * shift[o]   (vj becomes the layer output)
__global__ __launch_bounds__(256)
void finalize_k(float* __restrict__ vj, const float* __restrict__ rowsum,
                const float* __restrict__ shift)
{
    const size_t i4 = (size_t)blockIdx.x * 256 + threadIdx.x;  // float4 index
    const size_t e0 = i4 * 4;
    const int o = (int)(e0 & (DIM - 1));
    const int b = (int)(e0 >> 12);
    const float rs = rowsum[b];
    float4 v = ((float4*)vj)[i4];
    v.x += rs * shift[o + 0];
    v.y += rs * shift[o + 1];
    v.z += rs * shift[o + 2];
    v.w += rs * shift[o + 3];
    ((float4*)vj)[i4] = v;
}

// Wout[o,i] = W[o,i] + shift[o]   (pure bandwidth, float4)
__global__ __launch_bounds__(256)
void wupdate_k(const float* __restrict__ W, const float* __restrict__ shift,
               float* __restrict__ Wout)
{
    const size_t i4 = (size_t)blockIdx.x * 256 + threadIdx.x;  // float4 index
    const int o = (int)(i4 >> 10);                              // (i4*4)/DIM
    const float s = shift[o];
    float4 w = ((const float4*)W)[i4];
    w.x += s; w.y += s; w.z += s; w.w += s;
    ((float4*)Wout)[i4] = w;
}

extern "C" void kernel_launch(void* const* d_in, const int* in_sizes, int n_in,
                              void* d_out, int out_size, void* d_ws, size_t ws_size,
                              hipStream_t stream)
{
    (void)in_sizes; (void)n_in; (void)out_size; (void)ws_size;

    const float* x  = (const float*)d_in[0];
    const float* W1 = (const float*)d_in[1];
    const float* b1 = (const float*)d_in[2];
    const float* W2 = (const float*)d_in[3];
    const float* b2 = (const float*)d_in[4];
    const float* W3 = (const float*)d_in[5];
    const float* b3 = (const float*)d_in[6];
    const int*   bn = (const int*)d_in[7];

    float* out   = (float*)d_out;                       // [B_ROWS, DIM]
    float* outW1 = out   + (size_t)B_ROWS * DIM;
    float* outW2 = outW1 + (size_t)DIM * DIM;
    float* outW3 = outW2 + (size_t)DIM * DIM;

    float* buf0    = (float*)d_ws;                      // activation ping
    float* buf1    = buf0 + (size_t)B_ROWS * DIM;       // activation pong
    float* shift   = buf1 + (size_t)B_ROWS * DIM;       // [DIM]
    float* rsum    = shift + DIM;                       // [B_ROWS]
    float* partial = rsum + B_ROWS;                     // [8, DIM]

    struct Layer {
        const float* vi; const float* W; const float* b;
        float* vj; float* Wout;
    };
    const Layer L[3] = {
        { x,    W1, b1, buf0, outW1 },
        { buf0, W2, b2, buf1, outW2 },
        { buf1, W3, b3, out,  outW3 },
    };

    const dim3 blk(256);
    const dim3 gGemm(DIM / 128, B_ROWS / 128);
    const dim3 gColP(DIM / 256, 8);
    const int  gColF = DIM / 256;
    const int  gRow  = B_ROWS;
    const int  gFin  = (B_ROWS * DIM / 4) / 256;
    const int  gWupd = (int)(((size_t)DIM * DIM / 4) / 256);

    for (int l = 0; l < 3; ++l) {
        gemm_relu_wmma<<<gGemm, blk, 0, stream>>>(L[l].vi, L[l].W, L[l].b, L[l].vj);
        colsum_part   <<<gColP, blk, 0, stream>>>(L[l].vj, partial);
        colsum_final  <<<gColF, blk, 0, stream>>>(partial, shift, bn);
        rowsum_k      <<<gRow,  blk, 0, stream>>>(L[l].vi, rsum);
        finalize_k    <<<gFin,  blk, 0, stream>>>(L[l].vj, rsum, shift);
        wupdate_k     <<<gWupd, blk, 0, stream>>>(L[l].W, shift, L[l].Wout);
    }
}